// BasicSampler_79817672229371
// MI455X (gfx1250) — compile-verified
//
#include <hip/hip_runtime.h>

#define VOCAB   50257
#define TOPK    50
#define NTHR    1024
#define TOPP    0.9f
#define INVTEMP (1.0f / 0.9f)
#define NCHUNK  ((VOCAB - 1) / 4)   // 12564 exact 16B chunks covering elements 1..50256

#if defined(__HIP_DEVICE_COMPILE__) && __has_builtin(__builtin_amdgcn_global_load_async_to_lds_b32)
#define HAVE_ASYNC_LDS 1
#else
#define HAVE_ASYNC_LDS 0
#endif

typedef int v4i __attribute__((ext_vector_type(4)));
typedef __attribute__((address_space(1))) int  gint_t;   // global dword
typedef __attribute__((address_space(3))) int  lint_t;   // LDS dword
typedef __attribute__((address_space(1))) v4i  gv4i_t;   // global 4-dword
typedef __attribute__((address_space(3))) v4i  lv4i_t;   // LDS 4-dword

// Map float to an order-preserving unsigned key, pack with ~index so that
// u64 max() == (larger value, then smaller index) — matches top_k tie-break.
__device__ __forceinline__ unsigned long long pack_val_idx(float f, unsigned i) {
    unsigned u = __float_as_uint(f);
    unsigned key = (u & 0x80000000u) ? ~u : (u | 0x80000000u);
    return ((unsigned long long)key << 32) | (unsigned long long)(~i);
}

__device__ __forceinline__ float unpack_val(unsigned long long w) {
    unsigned key = (unsigned)(w >> 32);
    unsigned u = (key & 0x80000000u) ? (key & 0x7FFFFFFFu) : ~key;
    return __uint_as_float(u);
}

__device__ __forceinline__ unsigned long long shfl_down_u64(unsigned long long v, int d) {
    unsigned lo = (unsigned)(v & 0xFFFFFFFFull);
    unsigned hi = (unsigned)(v >> 32);
    lo = __shfl_down(lo, d, 32);   // wave32 on gfx1250
    hi = __shfl_down(hi, d, 32);
    return ((unsigned long long)hi << 32) | (unsigned long long)lo;
}

// Deterministic uniform in [0,1) from (key=42, batch) — splitmix64 style.
__device__ __forceinline__ float uniform01(unsigned b) {
    unsigned long long x = 0x2545F4914F6CDD1Dull ^ (42ull << 32) ^ (unsigned long long)(b + 1u);
    x *= 0x9E3779B97F4A7C15ull;
    x ^= x >> 30; x *= 0xBF58476D1CE4E5B9ull;
    x ^= x >> 27; x *= 0x94D049BB133111EBull;
    x ^= x >> 31;
    return (float)(x >> 40) * (1.0f / 16777216.0f);
}

// Dynamic LDS: 3-float front pad so vocab element 1 sits on a 16B LDS boundary
// (global element 1 is 16B-aligned after peeling element 0). ~201 KB of the
// CDNA5 WGP's 320 KB LDS — the whole row lives on-chip.
extern __shared__ __attribute__((aligned(16))) float s_row[];
#define ROW(i) s_row[(i) + 3]

__global__ __launch_bounds__(NTHR)
void topk_topp_sample_kernel(const float* __restrict__ logits, float* __restrict__ out, int B) {
    const int b    = blockIdx.x;
    const int tid  = threadIdx.x;
    const int lane = tid & 31;
    const int wid  = tid >> 5;

    __shared__ unsigned long long s_warp[NTHR / 32];
    __shared__ unsigned long long s_top[TOPK];
    __shared__ unsigned long long s_win;

    // Last position of row b: logits[b, 7, :]
    const float* src = logits + ((size_t)b * 8 + 7) * (size_t)VOCAB;

    // ---- Stage 1: single HBM pass, row -> LDS via wide async DMA (ASYNCcnt) ----
#if HAVE_ASYNC_LDS
    if (tid == 0) {   // peel unaligned leading element
        __builtin_amdgcn_global_load_async_to_lds_b32(
            (gint_t*)src, (lint_t*)(&ROW(0)), 0, 0);
    }
    for (int c = tid; c < NCHUNK; c += NTHR) {   // 16B-aligned b128 DMA for the rest
        __builtin_amdgcn_global_load_async_to_lds_b128(
            (gv4i_t*)(src + 1 + 4 * c),
            (lv4i_t*)(&ROW(1 + 4 * c)),
            0, 0);
    }
#if __has_builtin(__builtin_amdgcn_s_wait_asynccnt)
    __builtin_amdgcn_s_wait_asynccnt(0);
#else
    asm volatile("s_wait_asynccnt 0" ::: "memory");
#endif
#else
    for (int i = tid; i < VOCAB; i += NTHR) ROW(i) = src[i];
#endif
    __syncthreads();

    // ---- Stage 2: per-thread packed local max over strided slice ----
    unsigned long long loc = 0ull;
    for (int i = tid; i < VOCAB; i += NTHR) {
        unsigned long long p = pack_val_idx(ROW(i), (unsigned)i);
        loc = (p > loc) ? p : loc;
    }

    // ---- Stage 3: extract top-50 (argmax, remove, owner rescans its slice) ----
    for (int r = 0; r < TOPK; ++r) {
        unsigned long long w = loc;
        for (int off = 16; off > 0; off >>= 1) {
            unsigned long long o = shfl_down_u64(w, off);
            w = (o > w) ? o : w;
        }
        if (lane == 0) s_warp[wid] = w;
        __syncthreads();
        if (wid == 0) {
            unsigned long long ww = s_warp[lane];
            for (int off = 16; off > 0; off >>= 1) {
                unsigned long long o = shfl_down_u64(ww, off);
                ww = (o > ww) ? o : ww;
            }
            if (lane == 0) { s_win = ww; s_top[r] = ww; }
        }
        __syncthreads();
        unsigned long long win = s_win;
        unsigned idx = ~(unsigned)(win & 0xFFFFFFFFull);
        if (tid == (int)(idx % NTHR)) {
            ROW(idx) = -INFINITY;   // remove winner; only owner touches/rescans
            unsigned long long nl = 0ull;
            for (int i = tid; i < VOCAB; i += NTHR) {
                unsigned long long p = pack_val_idx(ROW(i), (unsigned)i);
                nl = (p > nl) ? p : nl;
            }
            loc = nl;
        }
        // next iteration's barriers order s_win/s_warp reuse; no extra barrier needed
    }

    // ---- Stage 4: softmax + top-p mask + inverse-CDF sample (thread 0, 50 elems) ----
    if (tid == 0) {
        float    p[TOPK];
        unsigned ix[TOPK];
        for (int r = 0; r < TOPK; ++r) {
            unsigned long long w = s_top[r];
            p[r]  = unpack_val(w) * INVTEMP;          // temperature applied post-selection
            ix[r] = ~(unsigned)(w & 0xFFFFFFFFull);
        }
        const float m = p[0];                          // sorted descending -> p[0] is max
        float sum = 0.0f;
        for (int r = 0; r < TOPK; ++r) { p[r] = __expf(p[r] - m); sum += p[r]; }
        // keep r iff r==0 or (normalized prefix mass before r) <= TOPP
        float prefix = 0.0f, keepMass = 0.0f;
        int nkeep = 0;
        for (int r = 0; r < TOPK; ++r) {
            bool keep = (r == 0) || (prefix <= TOPP * sum);
            prefix += p[r];
            if (!keep) break;
            keepMass += p[r];
            nkeep = r + 1;
        }
        const float target = uniform01((unsigned)b) * keepMass;
        float acc = 0.0f;
        int pick = nkeep - 1;
        for (int r = 0; r < nkeep; ++r) {
            acc += p[r];
            if (acc > target) { pick = r; break; }
        }
        out[b] = (float)ix[pick];
    }
}

extern "C" void kernel_launch(void* const* d_in, const int* in_sizes, int n_in,
                              void* d_out, int out_size, void* d_ws, size_t ws_size,
                              hipStream_t stream) {
    (void)n_in; (void)out_size; (void)d_ws; (void)ws_size;
    const float* logits = (const float*)d_in[0];
    float* out = (float*)d_out;
    const int B = in_sizes[0] / (8 * VOCAB);   // 64
    if (B <= 0) return;
    const size_t smem = (size_t)(VOCAB + 3) * sizeof(float);   // 201,040 B dynamic LDS
    topk_topp_sample_kernel<<<B, NTHR, smem, stream>>>(logits, out, B);
}